// ConstellationUnmapper_60524679135751
// MI455X (gfx1250) — compile-verified
//
#include <hip/hip_runtime.h>
#include <stdint.h>

// QAM64 / general-constellation nearest-neighbor demapper for MI455X (gfx1250).
//
// Roofline: 8 MB in + 4 MB out @ 23.3 TB/s ~= 0.52 us. For a uniform QAM grid
// (the reference's case, detected at runtime from the constellation input),
// nearest-neighbor is closed-form lattice quantization (~12 VALU/symbol), and
// the kernel is HBM-bound; data is staged through the CDNA5 async path
// (GLOBAL_LOAD_ASYNC_TO_LDS_B128 + S_WAIT_ASYNCCNT, double-buffered per wave).
//
// For a non-uniform constellation the kernel falls back to an exact 64-way
// argmin built on V_WMMA_F32_16X16X4_F32: maximize s = c.x - 0.5|c|^2 (bias in
// the C accumulator), then reduce d^2 = |x|^2 - 2s with the index packed into
// the low mantissa bits (v_min_u32 tree + one shfl_xor(16) across M-halves),
// which also reproduces argmin's lowest-index tie-break.

typedef __attribute__((ext_vector_type(2))) float v2f;
typedef __attribute__((ext_vector_type(8))) float v8f;

#define WAVES_PER_BLOCK 8
#define BLOCK_THREADS   256           // 8 wave32s
#define CHUNK           128           // symbols per wave per pipeline stage (32 lanes x 4)
#define BUF_BYTES       (2 * CHUNK * 4)     // one buffer: I[128] + Q[128] floats = 1 KB
#define LDS_FLOATS_PER_WAVE (2 * 2 * CHUNK) // double-buffered

__device__ __forceinline__ void issue_async_chunk(const float* __restrict__ xI,
                                                  const float* __restrict__ xQ,
                                                  int symBase, int lane,
                                                  uint32_t ldsBufByte) {
  // Per-lane: copy 16B (4 floats) of I and 16B of Q into this wave's LDS buffer.
  uint32_t goff = (uint32_t)(symBase + lane * 4) * 4u; // byte offset within plane
  uint32_t ldsI = ldsBufByte + (uint32_t)lane * 16u;
  uint32_t ldsQ = ldsI + (uint32_t)(CHUNK * 4);
  asm volatile("global_load_async_to_lds_b128 %0, %1, %2 offset:0"
               :: "v"(ldsI), "v"(goff), "s"(xI) : "memory");
  asm volatile("global_load_async_to_lds_b128 %0, %1, %2 offset:0"
               :: "v"(ldsQ), "v"(goff), "s"(xQ) : "memory");
}

__device__ __forceinline__ int quant8(float x, float inv, float bias) {
  // Nearest level index on a uniform 8-level grid, clamped to [0,7].
  int k = __float2int_rn(fmaf(x, inv, bias));
  k = (k < 0) ? 0 : k;
  k = (k > 7) ? 7 : k;
  return k;
}

extern "C" __global__ void __launch_bounds__(BLOCK_THREADS)
qam64_demap_kernel(const float* __restrict__ x,      // [2, nSym]: I plane then Q plane
                   const float* __restrict__ cst,    // [2, 64]
                   float* __restrict__ out,          // [nSym] indices as float
                   int nSym) {
  __shared__ float lds[WAVES_PER_BLOCK * LDS_FLOATS_PER_WAVE]; // 16 KB
  const int lane = threadIdx.x & 31;
  const int wave = threadIdx.x >> 5;

  const float* xI = x;
  const float* xQ = x + nSym;

  const int gw      = blockIdx.x * WAVES_PER_BLOCK + wave;
  const int gstride = gridDim.x * WAVES_PER_BLOCK;

  // Candidate lattice parameters. m = 8*i + q; cst[m] = L_I[m>>3], cst[64+m] = L_Q[m&7].
  const float minI  = cst[0];
  const float stepI = cst[8]  - cst[0];
  const float minQ  = cst[64];
  const float stepQ = cst[65] - cst[64];

  // Wave-uniform check: is the constellation a uniform I-major 8x8 grid?
  bool uniformGrid = (stepI != 0.0f) && (stepQ != 0.0f);
  {
    const float tolI = fabsf(stepI) * 0.015625f; // step/64
    const float tolQ = fabsf(stepQ) * 0.015625f;
#pragma unroll 1
    for (int m = 0; (m < 64) && uniformGrid; ++m) {
      float pi = fmaf((float)(m >> 3), stepI, minI);
      float pq = fmaf((float)(m & 7), stepQ, minQ);
      uniformGrid = (fabsf(cst[m] - pi) <= tolI) && (fabsf(cst[64 + m] - pq) <= tolQ);
    }
  }

  if (uniformGrid) {
    // ---------- Fast path: lattice quantization, async global->LDS pipeline ----------
    const float invI  = 1.0f / stepI;
    const float invQ  = 1.0f / stepQ;
    const float biasI = -minI * invI;
    const float biasQ = -minQ * invQ;

    float*   waveLds     = &lds[wave * LDS_FLOATS_PER_WAVE];
    uint32_t waveLdsByte = (uint32_t)(uintptr_t)waveLds; // low 32 bits of generic ptr = LDS offset

    const int nChunks = nSym / CHUNK;
    int buf = 0;
    if (gw < nChunks) {
      issue_async_chunk(xI, xQ, gw * CHUNK, lane, waveLdsByte); // prime buffer 0
    }

    for (int c = gw; c < nChunks; c += gstride) {
      const int cn = c + gstride;
      if (cn < nChunks) {
        issue_async_chunk(xI, xQ, cn * CHUNK, lane,
                          waveLdsByte + (uint32_t)((buf ^ 1) * BUF_BYTES));
        // 2 newest async ops may still be in flight; chunk c's 2 are done
        // (async loads complete in order).
        asm volatile("s_wait_asynccnt 0x2" ::: "memory");
      } else {
        asm volatile("s_wait_asynccnt 0x0" ::: "memory");
      }

      const float* bufp = waveLds + buf * (2 * CHUNK);
      const float4 vi = *(const float4*)(bufp + lane * 4);         // ds_load_b128
      const float4 vq = *(const float4*)(bufp + CHUNK + lane * 4); // ds_load_b128

      float4 r;
      {
        int i0 = quant8(vi.x, invI, biasI), q0 = quant8(vq.x, invQ, biasQ);
        int i1 = quant8(vi.y, invI, biasI), q1 = quant8(vq.y, invQ, biasQ);
        int i2 = quant8(vi.z, invI, biasI), q2 = quant8(vq.z, invQ, biasQ);
        int i3 = quant8(vi.w, invI, biasI), q3 = quant8(vq.w, invQ, biasQ);
        r.x = (float)((i0 << 3) | q0);
        r.y = (float)((i1 << 3) | q1);
        r.z = (float)((i2 << 3) | q2);
        r.w = (float)((i3 << 3) | q3);
      }
      *(float4*)(out + c * CHUNK + lane * 4) = r;                  // global_store_b128
      buf ^= 1;
    }

    // Tail (nSym % CHUNK): bounds-checked scalar quantization; cold at N=1M.
    const int tail = nChunks * CHUNK;
    for (int i = tail + (int)(blockIdx.x * blockDim.x + threadIdx.x); i < nSym;
         i += (int)(gridDim.x * blockDim.x)) {
      int ii = quant8(xI[i], invI, biasI);
      int qq = quant8(xQ[i], invQ, biasQ);
      out[i] = (float)((ii << 3) | qq);
    }
  } else {
    // ---------- General path: exact 64-way argmin via V_WMMA_F32_16X16X4_F32 ----------
    const int mlan = lane & 15;
    const int hi   = lane >> 4; // which M-half this lane's D registers cover

    // A tiles (resident, reused every iteration): A[m][K]: K0=cI, K1=cQ, K2=K3=0.
    // 32-bit A 16x4 layout: lanes 0-15 hold K=0,1; lanes 16-31 hold K=2,3.
    v2f a[4];
    // C bias tiles: C[m][n] = -0.5*|c_m|^2 (independent of n).
    // 32-bit D/C 16x16 layout: VGPR r, lane -> (M = r + 8*hi, N = lane%16).
    v8f cb[4];
#pragma unroll
    for (int t = 0; t < 4; ++t) {
      float ci = cst[t * 16 + mlan];
      float cq = cst[64 + t * 16 + mlan];
      a[t].x = hi ? 0.0f : ci;
      a[t].y = hi ? 0.0f : cq;
#pragma unroll
      for (int r = 0; r < 8; ++r) {
        int m = t * 16 + hi * 8 + r;
        float bi = cst[m], bq = cst[64 + m];
        cb[t][r] = -0.5f * fmaf(bi, bi, bq * bq);
      }
    }

    // 16 symbols per wave-iteration. Wave-uniform trip count; EXEC all-1s at WMMA.
    const int nBlk16 = nSym / 16;
    for (int blk = gw; blk < nBlk16; blk += gstride) {
      const int n = blk * 16 + mlan;
      float xi = xI[n], xq = xQ[n]; // both halves read the same 16 symbols
      v2f b;                        // B 4x16: K0=xI, K1=xQ on lanes 0-15; K2,K3=0
      b.x = hi ? 0.0f : xi;
      b.y = hi ? 0.0f : xq;
      float xx = fmaf(xi, xi, xq * xq);

      uint32_t best = 0xFFFFFFFFu;
#pragma unroll
      for (int t = 0; t < 4; ++t) {
        v8f d = __builtin_amdgcn_wmma_f32_16x16x4_f32(
            /*neg_a=*/false, a[t], /*neg_b=*/false, b,
            /*c_mod=*/(short)0, cb[t], /*reuse_a=*/false, /*reuse_b=*/false);
#pragma unroll
        for (int r = 0; r < 8; ++r) {
          // d2 = |x|^2 - 2*s >= 0; clamp fp noise so bit pattern is orderable.
          float d2 = fmaxf(fmaf(-2.0f, d[r], xx), 0.0f);
          uint32_t key = (__float_as_uint(d2) & 0xFFFFFFC0u) |
                         (uint32_t)(t * 16 + hi * 8 + r);
          best = (key < best) ? key : best;
        }
      }
      // Combine the two M-halves (lane L <-> L^16); u32 min keeps lowest index on ties.
      uint32_t other = (uint32_t)__shfl_xor((int)best, 16, 32);
      best = (other < best) ? other : best;
      if (lane < 16) out[n] = (float)(best & 63u);
    }

    // Tail (nSym % 16): scalar brute force; cold.
    const int tail = nBlk16 * 16;
    for (int i = tail + (int)(blockIdx.x * blockDim.x + threadIdx.x); i < nSym;
         i += (int)(gridDim.x * blockDim.x)) {
      float xi = xI[i], xq = xQ[i];
      float bd = 3.402823466e+38f;
      int bm = 0;
#pragma unroll 1
      for (int m = 0; m < 64; ++m) {
        float di = xi - cst[m], dq = xq - cst[64 + m];
        float d2 = fmaf(di, di, dq * dq);
        if (d2 < bd) { bd = d2; bm = m; }
      }
      out[i] = (float)bm;
    }
  }
}

extern "C" void kernel_launch(void* const* d_in, const int* in_sizes, int n_in,
                              void* d_out, int out_size, void* d_ws, size_t ws_size,
                              hipStream_t stream) {
  (void)n_in; (void)d_ws; (void)ws_size; (void)out_size;
  const float* x   = (const float*)d_in[0];   // [1,1,2,N] float32
  const float* cst = (const float*)d_in[1];   // [2,64]   float32
  float* out = (float*)d_out;                 // N indices (stored as float32)
  const int nSym = in_sizes[0] / 2;

  int nChunks = nSym / CHUNK;
  int blocks  = (nChunks + WAVES_PER_BLOCK - 1) / WAVES_PER_BLOCK;
  if (blocks > 512) blocks = 512;   // 512 blocks x 8 waves: 2 chunks/wave at N=1M
  if (blocks < 1)  blocks = 1;

  hipLaunchKernelGGL(qam64_demap_kernel, dim3(blocks), dim3(BLOCK_THREADS), 0, stream,
                     x, cst, out, nSym);
}